// MultiHeadAttention_44830868636192
// MI455X (gfx1250) — compile-verified
//
#include <hip/hip_runtime.h>

typedef __attribute__((ext_vector_type(2))) float v2f;
typedef __attribute__((ext_vector_type(8))) float v8f;

#define D_MODEL 1024
#define NUM_HEADS 16
#define DEPTH 64
#define BATCH 2
#define TQ 2048
#define TK 2048
#define KCHUNK 128

// ---- CDNA5 async global->LDS path (ASYNCcnt), with portable fallback -------
#if defined(__gfx1250__) &&                                                    \
    __has_builtin(__builtin_amdgcn_global_load_async_to_lds_b32) &&            \
    __has_builtin(__builtin_amdgcn_s_wait_asynccnt)
#define USE_ASYNC 1
#else
#define USE_ASYNC 0
#endif

#if USE_ASYNC
typedef __attribute__((address_space(1))) int as1_int;   // global (prints as __device__ int*)
typedef __attribute__((address_space(3))) int as3_int;   // LDS
#endif

static __device__ __forceinline__ void async_copy_f32(const float* g, float* l) {
#if USE_ASYNC
  // Per-lane global addr -> per-lane LDS addr; tracked by ASYNCcnt.
  // AS3 pointers are 32-bit; low 32 bits of a generic LDS pointer are the LDS
  // offset (aperture rule), so integer truncation is the portable cast.
  __builtin_amdgcn_global_load_async_to_lds_b32(
      (as1_int*)(unsigned long long)g,
      (as3_int*)(unsigned int)(unsigned long long)l, 0, 0);
#else
  *l = *g;
#endif
}

static __device__ __forceinline__ void wait_async_all() {
#if USE_ASYNC
  __builtin_amdgcn_s_wait_asynccnt(0);
#endif
}

// D = A(16x4, f32) x B(4x16, f32) + C(16x16, f32)  -- exact fp32 matrix pipe
static __device__ __forceinline__ v8f wmma_f32(v2f a, v2f b, v8f c) {
  return __builtin_amdgcn_wmma_f32_16x16x4_f32(false, a, false, b, (short)0, c,
                                               false, false);
}

// ---------------------------------------------------------------------------
// GEMM: Y = (X @ W + bias) * scale. Block = 8 waves = 128 rows x 16 cols.
// The shared 16-wide W strip is staged async into LDS in k-pair-major layout
// ([(k>>1)][col][k&1]) so each B fragment is one conflict-free ds_read_b64
// (grp0 lanes -> banks 0-31, grp1 lanes -> banks 32-63). Double buffered.
// ---------------------------------------------------------------------------
__global__ __launch_bounds__(256)
void gemm_proj(const float* __restrict__ X, const float* __restrict__ W,
               const float* __restrict__ bias, float* __restrict__ Out,
               float scale, int split_heads) {
  __shared__ float bs[2][(KCHUNK / 2) * 32];  // 2 x 8 KB

  const int tid  = threadIdx.x;
  const int lane = tid & 31;
  const int wave = tid >> 5;
  const int r    = lane & 15;   // A row within tile == B/C column within tile
  const int grp  = lane >> 4;   // K-pair / M-half selector

  const int row0 = (blockIdx.y * 8 + wave) * 16;
  const int col0 = blockIdx.x * 16;

  const float* Arow = X + (size_t)(row0 + r) * D_MODEL;

  // stage W[k0 .. k0+KCHUNK) x [col0 .. col0+16) into bs[buf]
  auto stage = [&](int k0, int buf) {
#pragma unroll
    for (int i = 0; i < (KCHUNK * 16) / 256; ++i) {
      const int f = tid + i * 256;          // coalesced 64B row segments
      const int k = f >> 4, c = f & 15;
      async_copy_f32(W + (size_t)(k0 + k) * D_MODEL + col0 + c,
                     &bs[buf][(k >> 1) * 32 + c * 2 + (k & 1)]);
    }
  };

  stage(0, 0);
  v8f acc = {};
  for (int kc = 0; kc < D_MODEL / KCHUNK; ++kc) {
    const int buf = kc & 1;
    wait_async_all();       // each wave drains its own ASYNCcnt ...
    __syncthreads();        // ... then the barrier makes all LDS writes visible
    if (kc + 1 < D_MODEL / KCHUNK) stage((kc + 1) * KCHUNK, buf ^ 1);

    const float* bp = bs[buf];
    const float* ap = Arow + kc * KCHUNK;
#pragma unroll 8
    for (int kk = 0; kk < KCHUNK; kk += 4) {
      v2f a = *(const v2f*)(ap + kk + 2 * grp);                 // A[r][ka..ka+1]
      v2f b = *(const v2f*)(bp + (kk / 2 + grp) * 32 + r * 2);  // B[ka..ka+1][col]
      acc = wmma_f32(a, b, acc);
    }
    __syncthreads();        // all waves done reading buf before it is restaged
  }

  const int   n  = col0 + r;
  const float bb = bias[n];
#pragma unroll
  for (int i = 0; i < 8; ++i) {
    const int m = row0 + i + 8 * grp;           // C layout: M = i + 8*grp
    const float v = (acc[i] + bb) * scale;
    if (split_heads) {
      const int b_ = m >> 11, t = m & 2047;
      const int h  = n >> 6,  d = n & 63;
      Out[(((size_t)b_ * NUM_HEADS + h) * TQ + t) * DEPTH + d] = v;
    } else {
      Out[(size_t)m * D_MODEL + n] = v;
    }
  }
}

// ---------------------------------------------------------------------------
// Fused attention for one (b*16+h, 16-query block):
//   logits (WMMA) -> mask -> 128 KB LDS slab -> softmax in LDS -> probs to
//   d_out -> PV (WMMA, split-K over 2 waves) -> concat-layout output.
// ---------------------------------------------------------------------------
__global__ __launch_bounds__(256)
void attention_kernel(const float* __restrict__ Qh, const float* __restrict__ Kh,
                      const float* __restrict__ Vh, const float* __restrict__ mask,
                      float* __restrict__ attn_out,     // (B,H,TQ,TK)
                      float* __restrict__ concat_out) { // (B,TQ,D_MODEL)
  extern __shared__ float smem[];
  float* probs = smem;                   // 16*2048 floats (128 KB)
  float* qs    = probs + 16 * TK;        // 16*64
  float* red   = qs + 16 * DEPTH;        // 16*16 partials
  float* rowv  = red + 16 * 16;          // 16 (row max / inv-sum)
  float* avls  = rowv + 16;              // 8*256 split-K partial C tiles

  const int tid  = threadIdx.x;
  const int lane = tid & 31;
  const int wave = tid >> 5;
  const int r    = lane & 15;
  const int grp  = lane >> 4;

  const int bh = blockIdx.y;             // b*16 + h
  const int q0 = blockIdx.x * 16;

  const float* Qb = Qh + ((size_t)bh * TQ + q0) * DEPTH;
  const float* Kb = Kh + (size_t)bh * TK * DEPTH;
  const float* Vb = Vh + (size_t)bh * TK * DEPTH;

  for (int i = tid; i < 16 * DEPTH; i += 256) qs[i] = Qb[i];
  __syncthreads();

  // ---- logits: 128 key tiles, wave-strided ----
  for (int nt = wave; nt < TK / 16; nt += 8) {
    const int key = nt * 16 + r;
    const float* Bp = Kb + (size_t)key * DEPTH;  // B[k][n] = K[key][k] (contig)
    if (nt + 8 < TK / 16)                        // prefetch next tile's K rows
      __builtin_prefetch(Bp + (size_t)128 * DEPTH, 0, 1);
    v8f acc = {};
#pragma unroll
    for (int k = 0; k < DEPTH; k += 4) {
      const int ka = k + 2 * grp;
      v2f a = *(const v2f*)(qs + r * DEPTH + ka);
      v2f b = *(const v2f*)(Bp + ka);
      acc = wmma_f32(a, b, acc);
    }
#pragma unroll
    for (int i = 0; i < 8; ++i) {
      const int m  = i + 8 * grp;
      const int qi = q0 + m;
      const float mval = mask[(size_t)qi * TK + key];   // L2-resident
      probs[m * TK + nt * 16 + r] = acc[i] + mval * -1e9f;
    }
  }
  __syncthreads();

  // ---- softmax over each of the 16 rows (16 threads x 128 elems per row) ----
  const int srow   = tid >> 4;
  const int schunk = tid & 15;
  float* prow = probs + srow * TK + schunk * 128;

  float pmax = -3.0e38f;
  for (int j = 0; j < 128; ++j) pmax = fmaxf(pmax, prow[j]);
  red[srow * 16 + schunk] = pmax;
  __syncthreads();
  if (schunk == 0) {
    float m = red[srow * 16];
    for (int j = 1; j < 16; ++j) m = fmaxf(m, red[srow * 16 + j]);
    rowv[srow] = m;
  }
  __syncthreads();
  const float rmax = rowv[srow];
  float psum = 0.f;
  for (int j = 0; j < 128; ++j) {
    const float e = __expf(prow[j] - rmax);
    prow[j] = e;
    psum += e;
  }
  red[srow * 16 + schunk] = psum;
  __syncthreads();
  if (schunk == 0) {
    float s = 0.f;
    for (int j = 0; j < 16; ++j) s += red[srow * 16 + j];
    rowv[srow] = 1.0f / s;
  }
  __syncthreads();
  const float rinv = rowv[srow];
  {
    float* aout = attn_out + ((size_t)bh * TQ + (q0 + srow)) * TK + schunk * 128;
    for (int j = 0; j < 128; ++j) {
      const float p = prow[j] * rinv;
      prow[j] = p;          // normalized probs stay in LDS for PV
      aout[j] = p;          // the mandatory 536 MB output stream
    }
  }
  __syncthreads();

  // ---- PV: 4 depth tiles x split-K(2) across 8 waves ----
  {
    const int nt  = wave & 3;
    const int kh  = wave >> 2;
    const int col = nt * 16 + r;
    v8f acc = {};
    for (int k = kh * (TK / 2); k < (kh + 1) * (TK / 2); k += 4) {
      const int ka = k + 2 * grp;
      if ((k & 63) == 0)                 // prefetch V rows one chunk ahead
        __builtin_prefetch(Vb + (size_t)(ka + 64) * DEPTH + col, 0, 1);
      v2f a = *(const v2f*)(probs + r * TK + ka);
      v2f b;
      b.x = Vb[(size_t)ka * DEPTH + col];
      b.y = Vb[(size_t)(ka + 1) * DEPTH + col];
      acc = wmma_f32(a, b, acc);
    }
#pragma unroll
    for (int i = 0; i < 8; ++i)
      avls[wave * 256 + (i + 8 * grp) * 16 + r] = acc[i];
  }
  __syncthreads();

  const int b_ = bh >> 4;
  const int h  = bh & 15;
  for (int e = tid; e < 4 * 256; e += 256) {
    const int tile = e >> 8;
    const int m = (e >> 4) & 15;
    const int n = e & 15;
    const float v = avls[tile * 256 + m * 16 + n] +
                    avls[(tile + 4) * 256 + m * 16 + n];
    concat_out[((size_t)b_ * TQ + (q0 + m)) * D_MODEL + h * DEPTH + tile * 16 + n] = v;
  }
}

// ---------------------------------------------------------------------------
extern "C" void kernel_launch(void* const* d_in, const int* in_sizes, int n_in,
                              void* d_out, int out_size, void* d_ws, size_t ws_size,
                              hipStream_t stream) {
  (void)in_sizes; (void)n_in; (void)out_size; (void)ws_size;

  const float* queries = (const float*)d_in[0];
  const float* keys    = (const float*)d_in[1];
  const float* values  = (const float*)d_in[2];
  const float* mask    = (const float*)d_in[3];
  const float* wq = (const float*)d_in[4];
  const float* bq = (const float*)d_in[5];
  const float* wk = (const float*)d_in[6];
  const float* bk = (const float*)d_in[7];
  const float* wv = (const float*)d_in[8];
  const float* bv = (const float*)d_in[9];
  const float* wo = (const float*)d_in[10];
  const float* bo = (const float*)d_in[11];

  float* out  = (float*)d_out;                        // (B,TQ,D_MODEL)
  float* attn = out + (size_t)BATCH * TQ * D_MODEL;   // (B,H,TQ,TK)

  const size_t hsz = (size_t)BATCH * NUM_HEADS * TQ * DEPTH;  // 4 Mi floats
  float* Qh     = (float*)d_ws;
  float* Kh     = Qh + hsz;
  float* Vh     = Kh + hsz;
  float* concat = Vh + hsz;

  dim3 blk(256);
  dim3 gproj(D_MODEL / 16, (BATCH * TQ) / (16 * 8));  // (64, 32)

  // Q/K/V projections; 1/sqrt(depth)=0.125 folded into Q (incl. bias).
  gemm_proj<<<gproj, blk, 0, stream>>>(queries, wq, bq, Qh, 0.125f, 1);
  gemm_proj<<<gproj, blk, 0, stream>>>(keys,    wk, bk, Kh, 1.0f,   1);
  gemm_proj<<<gproj, blk, 0, stream>>>(values,  wv, bv, Vh, 1.0f,   1);

  dim3 gattn(TQ / 16, BATCH * NUM_HEADS);             // (128, 32)
  const size_t smem =
      (size_t)(16 * TK + 16 * DEPTH + 16 * 16 + 16 + 8 * 256) * sizeof(float);
  attention_kernel<<<gattn, blk, smem, stream>>>(Qh, Kh, Vh, mask, attn, concat);

  gemm_proj<<<gproj, blk, 0, stream>>>(concat, wo, bo, out, 1.0f, 0);
}